// SpikeFP8Adder_Spatial_76312978916089
// MI455X (gfx1250) — compile-verified
//
#include <hip/hip_runtime.h>
#include <stdint.h>

// ---------------------------------------------------------------------------
// SpikeFP8Adder_Spatial on MI455X (gfx1250)
//
// The reference is a boolean circuit over {0.0,1.0} floats: an E4M3 FP8 adder
// built from XOR/AND/OR/MUX gates. We evaluate the *exact* circuit with
// integer bit ops per element (pack 8 floats -> byte, emulate, unpack).
//
// Roofline: 402 MB total traffic / 23.3 TB/s  ~= 17 us. Memory bound.
// Data path: CDNA5 async global->LDS copies (ASYNCcnt) with a double-buffered
// per-wave pipeline, 128-bit transactions end to end.
// ---------------------------------------------------------------------------

#define TPB 256
// per-thread LDS slot: 8 A-words + 8 B-words = 64 bytes; two buffers.
#define WORDS_PER_THREAD 16

// pack 8 wire-floats (0.0f / 1.0f, MSB-first) into a byte (bit7 = index 0)
__device__ __forceinline__ uint32_t pack8(uint4 lo, uint4 hi) {
  uint32_t v;
  v  = ((lo.x >> 23) & 1u) << 7;   // 1.0f = 0x3F800000 -> bit23 set
  v |= ((lo.y >> 23) & 1u) << 6;
  v |= ((lo.z >> 23) & 1u) << 5;
  v |= ((lo.w >> 23) & 1u) << 4;
  v |= ((hi.x >> 23) & 1u) << 3;
  v |= ((hi.y >> 23) & 1u) << 2;
  v |= ((hi.z >> 23) & 1u) << 1;
  v |= ((hi.w >> 23) & 1u);
  return v;
}

// Exact integer emulation of the reference gate network.
// Layout: bit7=sign, bits6..3=exp (MSB-first), bits2..0=mantissa (MSB-first).
__device__ __forceinline__ uint32_t fp8_add_bits(uint32_t va, uint32_t vb) {
  const uint32_t sa = (va >> 7) & 1u, ea = (va >> 3) & 15u, ma = va & 7u;
  const uint32_t sb = (vb >> 7) & 1u, eb = (vb >> 3) & 15u, mb = vb & 7u;

  // comparator4 -> sel = (ea >= eb); MUX swap to larger/smaller operand
  const bool sel = (ea >= eb);
  const uint32_t exp_l  = sel ? ea : eb, exp_s  = sel ? eb : ea;
  const uint32_t man_l  = sel ? ma : mb, man_s  = sel ? mb : ma;
  const uint32_t sign_l = sel ? sa : sb, sign_s = sel ? sb : sa;

  // subtract_bits(exp_l, exp_s): exp_l >= exp_s so plain difference
  const uint32_t diff = (exp_l - exp_s) & 15u;

  // hidden bit = OR of exponent bits; ext vector bit10=hid, bits9..7=man
  const uint32_t ext_l =  ((exp_l != 0u) ? 1024u : 0u) | (man_l << 7);
  uint32_t       ext_s = (((exp_s != 0u) ? 1024u : 0u) | (man_s << 7)) >> diff; // barrel_right12

  // 12-bit modular add / subtract (carry / borrow dropped)
  const uint32_t ssum = (ext_l + ext_s) & 0xFFFu;
  const uint32_t sdif = (ext_l - ext_s) & 0xFFFu;
  const uint32_t mant = (sign_l == sign_s) ? ssum : sdif;

  // top8 = mant bits 11..4; lzd8 (returns 7 for zero input -> inject bit24)
  const uint32_t top8 = (mant >> 4) & 0xFFu;
  const uint32_t lzc  = (uint32_t)__builtin_clz((top8 << 24) | (1u << 24));
  const uint32_t norm = (top8 << lzc) & 0xFFu;                 // barrel_left8

  // exp_new = increment4(subtract_bits(exp_l, lzc)) = (exp_l - lzc + 1) mod 16
  const uint32_t exp_new = (exp_l - lzc + 1u) & 15u;

  // result = [sign_l, exp_new(4), norm bits 6..4]
  return (sign_l << 7) | (exp_new << 3) | ((norm >> 4) & 7u);
}

__device__ __forceinline__ void unpack_store(float* __restrict__ O,
                                             size_t elem, uint32_t res) {
  float4 o0, o1;
  o0.x = (float)((res >> 7) & 1u);
  o0.y = (float)((res >> 6) & 1u);
  o0.z = (float)((res >> 5) & 1u);
  o0.w = (float)((res >> 4) & 1u);
  o1.x = (float)((res >> 3) & 1u);
  o1.y = (float)((res >> 2) & 1u);
  o1.z = (float)((res >> 1) & 1u);
  o1.w = (float)( res       & 1u);
  float4* op = (float4*)(O + elem * 8u);
  op[0] = o0;                                  // global_store_b128
  op[1] = o1;                                  // global_store_b128
}

// Issue async global->LDS copies for one tile (per thread: 32B A + 32B B).
// offset: applies to BOTH the LDS and global address, so two b128 per stream.
// s_wait_dscnt 0 first: the buffer we refill was read via ds_load last round.
__device__ __forceinline__ void issue_tile(const float* __restrict__ A,
                                           const float* __restrict__ B,
                                           int tile, int tid, uint32_t lds) {
  const uint64_t ga = (uint64_t)(uintptr_t)(A + (size_t)tile * (TPB * 8) + tid * 8);
  const uint64_t gb = (uint64_t)(uintptr_t)(B + (size_t)tile * (TPB * 8) + tid * 8);
  asm volatile(
      "s_wait_dscnt 0\n\t"
      "global_load_async_to_lds_b128 %0, %2, off\n\t"
      "global_load_async_to_lds_b128 %0, %2, off offset:16\n\t"
      "global_load_async_to_lds_b128 %1, %3, off\n\t"
      "global_load_async_to_lds_b128 %1, %3, off offset:16"
      :
      : "v"(lds), "v"(lds + 32u), "v"(ga), "v"(gb)
      : "memory");
}

__global__ __launch_bounds__(TPB) void spike_fp8_adder_kernel(
    const float* __restrict__ A, const float* __restrict__ B,
    float* __restrict__ O, int nElem, int nTiles) {
  __shared__ uint32_t smem[2 * TPB * WORDS_PER_THREAD];   // 32 KB
  const int tid = threadIdx.x;

  // raw LDS byte addresses of this thread's two buffers (low 32 bits of the
  // generic pointer are the wave-relative LDS offset on amdgcn)
  uint32_t ldsBuf[2];
  ldsBuf[0] = (uint32_t)(uintptr_t)&smem[0 * TPB * WORDS_PER_THREAD + tid * WORDS_PER_THREAD];
  ldsBuf[1] = (uint32_t)(uintptr_t)&smem[1 * TPB * WORDS_PER_THREAD + tid * WORDS_PER_THREAD];

  int tile = (int)blockIdx.x;
  if (tile < nTiles) {
    int buf = 0;
    issue_tile(A, B, tile, tid, ldsBuf[buf]);                 // prologue
    for (;;) {
      const int next = tile + (int)gridDim.x;
      if (next < nTiles) {
        issue_tile(A, B, next, tid, ldsBuf[buf ^ 1]);         // prefetch k+1
        // 4 outstanding = the copies just issued; tile k's data has landed
        asm volatile("s_wait_asynccnt 4" ::: "memory");
      } else {
        asm volatile("s_wait_asynccnt 0" ::: "memory");
      }

      // consume tile k from LDS (ds_load_b128 x4 per thread)
      const uint4* sp = (const uint4*)&smem[buf * TPB * WORDS_PER_THREAD +
                                           tid * WORDS_PER_THREAD];
      const uint4 a0 = sp[0], a1 = sp[1], b0 = sp[2], b1 = sp[3];
      const uint32_t res = fp8_add_bits(pack8(a0, a1), pack8(b0, b1));
      unpack_store(O, (size_t)tile * TPB + tid, res);

      if (next >= nTiles) break;
      tile = next;
      buf ^= 1;
    }
  }

  // tail (< TPB elements): direct global loads, block 0 only
  const int e = nTiles * TPB + tid;
  if (blockIdx.x == 0 && e < nElem) {
    const uint4* pa = (const uint4*)(A + (size_t)e * 8u);
    const uint4* pb = (const uint4*)(B + (size_t)e * 8u);
    const uint4 a0 = pa[0], a1 = pa[1], b0 = pb[0], b1 = pb[1];
    const uint32_t res = fp8_add_bits(pack8(a0, a1), pack8(b0, b1));
    unpack_store(O, (size_t)e, res);
  }
}

extern "C" void kernel_launch(void* const* d_in, const int* in_sizes, int n_in,
                              void* d_out, int out_size, void* d_ws, size_t ws_size,
                              hipStream_t stream) {
  const float* A = (const float*)d_in[0];
  const float* B = (const float*)d_in[1];
  float* O = (float*)d_out;

  const int nElem  = in_sizes[0] / 8;     // (N, 8) float32
  const int nTiles = nElem / TPB;

  int blocks = 2048;                      // 8 tiles/block at N = 4M
  if (blocks > nTiles) blocks = nTiles;
  if (blocks < 1) blocks = 1;

  spike_fp8_adder_kernel<<<blocks, TPB, 0, stream>>>(A, B, O, nElem, nTiles);
}